// GNNBlock_57114475102835
// MI455X (gfx1250) — compile-verified
//
#include <hip/hip_runtime.h>
#include <hip/hip_bf16.h>

// ---------------------------------------------------------------------------
// GraphSAGE 2-layer block for MI455X (gfx1250, wave32).
//  - Edge aggregation: one wave32 per edge, b128 gather + 4 f32 atomics/lane.
//  - Mean applied in-place (elementwise) so the GEMM A panel is a pure copy.
//  - GEMM: V_WMMA_F32_16X16X4_F32, 8 waves/block, 64x128 output tile/block,
//    A panel (64 x 256 = [mean_agg ; x]) staged in LDS via CDNA5 async
//    global->LDS loads (ASYNCcnt), B panel K-contiguous for b64 loads.
// ---------------------------------------------------------------------------

typedef __attribute__((ext_vector_type(2))) float v2f;
typedef __attribute__((ext_vector_type(8))) float v8f;

#define NNODES 50000
#define DFEAT  128
#define KTOT   256              // concat K: [agg_mean ; x]
#define CLAMPV 1000.0f
#define MSUB   4                // 4 x 16 = 64 rows per block
#define APAD   260              // 256+4: row stride mod 64 banks = 4 -> conflict-free

#if __has_builtin(__builtin_amdgcn_global_load_async_to_lds_b64)
#define HAS_ASYNC_LDS 1
// builtin expects: (AS1 v2i*, AS3 v2i*, imm offset, imm cpol)
typedef int v2i_vs __attribute__((vector_size(2 * sizeof(int))));
typedef __attribute__((address_space(1))) v2i_vs* gv2i_p;
typedef __attribute__((address_space(3))) v2i_vs* lv2i_p;
__device__ __forceinline__ void async_copy_b64(const void* g, void* l) {
    __builtin_amdgcn_global_load_async_to_lds_b64((gv2i_p)g, (lv2i_p)l, 0, 0);
}
__device__ __forceinline__ void async_wait0() {
#if __has_builtin(__builtin_amdgcn_s_wait_asynccnt)
    __builtin_amdgcn_s_wait_asynccnt(0);
#else
    asm volatile("s_wait_asynccnt 0x0" ::: "memory");
#endif
}
#endif

// ---------------- utility kernels ----------------

__global__ void sage_zero_kernel(float* __restrict__ p, int n) {
    int i = blockIdx.x * blockDim.x + threadIdx.x;
    int stride = gridDim.x * blockDim.x;
    for (; i < n; i += stride) p[i] = 0.0f;
}

__global__ void sage_clamp_kernel(const float* __restrict__ x,
                                  float* __restrict__ xc, int n) {
    int i = blockIdx.x * blockDim.x + threadIdx.x;
    int stride = gridDim.x * blockDim.x;
    for (; i < n; i += stride) {
        float v = x[i];
        xc[i] = fminf(fmaxf(v, -CLAMPV), CLAMPV);
    }
}

// Build transposed concat weight: WcatT[col][k], col in [0,128), k in [0,256)
__global__ void sage_wcatT_kernel(const float* __restrict__ Wl,
                                  const float* __restrict__ Wr,
                                  float* __restrict__ WcatT) {
    int i = blockIdx.x * blockDim.x + threadIdx.x;   // 0 .. 32767
    if (i >= DFEAT * KTOT) return;
    int col = i >> 8;
    int k   = i & 255;
    WcatT[i] = (k < DFEAT) ? Wl[k * DFEAT + col]
                           : Wr[(k - DFEAT) * DFEAT + col];
}

// In-place mean: agg[node][*] *= 1/max(deg[node],1). One float4 per thread.
__global__ void sage_mean_kernel(float* __restrict__ agg,
                                 const float* __restrict__ deg, int nNodes) {
    int i = blockIdx.x * blockDim.x + threadIdx.x;   // float4 index
    if (i >= nNodes * (DFEAT / 4)) return;
    int node = i >> 5;                               // 32 float4 per node
    float r = 1.0f / fmaxf(deg[node], 1.0f);
    float4* p = (float4*)agg + i;
    float4 v = *p;
    v.x *= r; v.y *= r; v.z *= r; v.w *= r;
    *p = v;
}

// ---------------- edge scatter: one wave32 per edge ----------------

__global__ __launch_bounds__(256)
void sage_scatter_kernel(const float* __restrict__ feat,
                         const int* __restrict__ src,
                         const int* __restrict__ dst,
                         float* __restrict__ agg,
                         float* __restrict__ deg, int nEdges) {
    int warp = (blockIdx.x * blockDim.x + threadIdx.x) >> 5;
    int lane = threadIdx.x & 31;
    if (warp >= nEdges) return;
    int s = src[warp];
    int d = dst[warp];
    const float* fs = feat + (long)s * DFEAT;
    float*       ad = agg  + (long)d * DFEAT;
    float4 v = *(const float4*)&fs[lane << 2];   // one b128 gather per lane
    float* a4 = ad + (lane << 2);
    atomicAdd(a4 + 0, v.x);
    atomicAdd(a4 + 1, v.y);
    atomicAdd(a4 + 2, v.z);
    atomicAdd(a4 + 3, v.w);
    if (lane == 0) atomicAdd(deg + d, 1.0f);
}

// ---------------- fused SAGE GEMM via WMMA f32 16x16x4 ----------------
// out[64x128 tile] = relu_clamp( [aggM ; x](64x256) @ W(256x128) + b )

__global__ __launch_bounds__(256)
void sage_gemm_wmma_kernel(const float* __restrict__ aggM,  // pre-meaned
                           const float* __restrict__ xin,
                           const float* __restrict__ WcatT, // [128][256]
                           const float* __restrict__ bias,
                           float* __restrict__ out, int nNodes) {
    __shared__ float As[64][APAD];     // 64 rows x 256 K (padded)

    const int tid  = threadIdx.x;
    const int lane = tid & 31;
    const int wave = tid >> 5;                 // 0..7 -> N tile
    const int m0   = blockIdx.x * 64;

    const int nsub = lane & 15;
    const int kb   = (lane >> 4) << 1;         // 0 or 2
    const int col  = (wave << 4) + nsub;       // 0..127

    // get the weight panel in flight while A is staged
    __builtin_prefetch(&WcatT[col * KTOT], 0, 0);

    // -------- A-panel staging: 64 rows x 256 K --------
    // Rows beyond nNodes are clamped (their garbage only feeds D rows that
    // are never stored), keeping the staging loop EXEC-uniform.
#ifdef HAS_ASYNC_LDS
    // CDNA5 async global->LDS copy, b64 granularity (8B-aligned rows).
#pragma unroll 4
    for (int it = 0; it < 32; ++it) {
        int flat = tid + (it << 8);            // 0..8191 (b64 units)
        int row  = flat >> 7;                  // 0..63
        int k    = (flat & 127) << 1;          // 0,2,...,254
        int node = min(m0 + row, nNodes - 1);
        const float* src = (k < DFEAT)
            ? (aggM + (long)node * DFEAT + k)
            : (xin  + (long)node * DFEAT + (k - DFEAT));
        async_copy_b64(src, &As[row][k]);
    }
    async_wait0();                             // own wave's ASYNCcnt -> 0
    __syncthreads();                           // cross-wave LDS visibility
#else
    // fallback: float4 global loads + two b64 LDS stores
#pragma unroll 4
    for (int it = 0; it < 16; ++it) {
        int flat = tid + (it << 8);            // 0..4095 (float4 units)
        int row  = flat >> 6;                  // 0..63
        int kq   = (flat & 63) << 2;           // 0,4,...,252
        int node = min(m0 + row, nNodes - 1);
        float4 v = (kq < DFEAT)
            ? *(const float4*)&aggM[(long)node * DFEAT + kq]
            : *(const float4*)&xin[(long)node * DFEAT + (kq - DFEAT)];
        *(v2f*)&As[row][kq]     = (v2f){v.x, v.y};
        *(v2f*)&As[row][kq + 2] = (v2f){v.z, v.w};
    }
    __syncthreads();
#endif

    // WMMA fragment indexing (f32 16x16x4 layout, ISA 7.12.2):
    //  A: lane -> M = lane&15, K pair base kb = 2*(lane>>4)
    //  B: lane -> N = lane&15, same kb
    //  D: VGPR r -> M = r + 8*(lane>>4), N = lane&15
    v8f acc[MSUB];
#pragma unroll
    for (int m = 0; m < MSUB; ++m) acc[m] = (v8f)(0.0f);

    const float* __restrict__ wcol = WcatT + (long)col * KTOT;

    for (int kk = 0; kk < KTOT / 4; ++kk) {    // 64 K-steps of 4
        const int k = (kk << 2) + kb;
        v2f b = *(const v2f*)&wcol[k];          // single b64, K-contiguous
#pragma unroll
        for (int m = 0; m < MSUB; ++m) {
            const int row = (m << 4) + nsub;
            v2f a = *(const v2f*)&As[row][k];   // 8B aligned (k even)
            acc[m] = __builtin_amdgcn_wmma_f32_16x16x4_f32(
                false, a, false, b, (short)0, acc[m], false, false);
        }
    }

    // epilogue: + bias, relu, clamp; block-uniform fast path (no guards)
    const float bv = bias[col];
    const int rbase = (lane >> 4) << 3;        // 0 or 8
    if (m0 + 64 <= nNodes) {
        float* obase = out + (long)m0 * DFEAT + col;
#pragma unroll
        for (int m = 0; m < MSUB; ++m) {
#pragma unroll
            for (int r = 0; r < 8; ++r) {
                int row = (m << 4) + rbase + r;
                float v = acc[m][r] + bv;
                v = fminf(fmaxf(v, 0.0f), CLAMPV);
                obase[(long)row * DFEAT] = v;
            }
        }
    } else {
#pragma unroll
        for (int m = 0; m < MSUB; ++m) {
#pragma unroll
            for (int r = 0; r < 8; ++r) {
                int row  = (m << 4) + rbase + r;
                int node = m0 + row;
                float v = acc[m][r] + bv;
                v = fminf(fmaxf(v, 0.0f), CLAMPV);
                if (node < nNodes) out[(long)node * DFEAT + col] = v;
            }
        }
    }
}

// ---------------- host launcher ----------------

extern "C" void kernel_launch(void* const* d_in, const int* in_sizes, int n_in,
                              void* d_out, int out_size, void* d_ws, size_t ws_size,
                              hipStream_t stream) {
    const float* x    = (const float*)d_in[0];
    const int*   ei   = (const int*)d_in[1];
    const float* Wl1  = (const float*)d_in[2];
    const float* b1   = (const float*)d_in[3];
    const float* Wr1  = (const float*)d_in[4];
    const float* Wl2  = (const float*)d_in[5];
    const float* b2   = (const float*)d_in[6];
    const float* Wr2  = (const float*)d_in[7];
    float* out = (float*)d_out;

    const int E  = in_sizes[1] / 2;
    const int ND = NNODES * DFEAT;
    const int src_off = 0, dst_off = E;

    // workspace layout (floats)
    float* ws     = (float*)d_ws;
    float* xc     = ws;                 // N*D clamped input
    float* h1     = ws + (size_t)ND;    // N*D layer-1 output
    float* agg    = ws + (size_t)2 * ND;
    float* dg     = ws + (size_t)3 * ND;
    float* WcatT1 = dg + NNODES;
    float* WcatT2 = WcatT1 + KTOT * DFEAT;

    const int TB = 256;
    const int gridZeroA = (ND + TB * 16 - 1) / (TB * 16);
    const int gridZeroD = (NNODES + TB - 1) / TB;
    const int gridClamp = (ND + TB * 8 - 1) / (TB * 8);
    const int gridWcat  = (KTOT * DFEAT + TB - 1) / TB;
    const int gridMean  = (NNODES * (DFEAT / 4) + TB - 1) / TB;
    const int gridScat  = (E + 7) / 8;                 // 8 edges (waves) / block
    const int gridGemm  = (NNODES + 63) / 64;

    // ---- prologue ----
    sage_zero_kernel<<<gridZeroA, TB, 0, stream>>>(agg, ND);
    sage_zero_kernel<<<gridZeroD, TB, 0, stream>>>(dg, NNODES);
    sage_clamp_kernel<<<gridClamp, TB, 0, stream>>>(x, xc, ND);
    sage_wcatT_kernel<<<gridWcat, TB, 0, stream>>>(Wl1, Wr1, WcatT1);
    sage_wcatT_kernel<<<gridWcat, TB, 0, stream>>>(Wl2, Wr2, WcatT2);

    // ---- layer 1 ----
    sage_scatter_kernel<<<gridScat, TB, 0, stream>>>(xc, ei + src_off, ei + dst_off,
                                                     agg, dg, E);
    sage_mean_kernel<<<gridMean, TB, 0, stream>>>(agg, dg, NNODES);
    sage_gemm_wmma_kernel<<<gridGemm, TB, 0, stream>>>(agg, xc, WcatT1, b1, h1,
                                                       NNODES);

    // ---- layer 2 ----
    sage_zero_kernel<<<gridZeroA, TB, 0, stream>>>(agg, ND);
    sage_zero_kernel<<<gridZeroD, TB, 0, stream>>>(dg, NNODES);
    sage_scatter_kernel<<<gridScat, TB, 0, stream>>>(h1, ei + src_off, ei + dst_off,
                                                     agg, dg, E);
    sage_mean_kernel<<<gridMean, TB, 0, stream>>>(agg, dg, NNODES);
    sage_gemm_wmma_kernel<<<gridGemm, TB, 0, stream>>>(agg, h1, WcatT2, b2, out,
                                                       NNODES);
}